// FPLPGCN_1168231104603
// MI455X (gfx1250) — compile-verified
//
#include <hip/hip_runtime.h>
#include <math.h>

// Problem constants (match reference setup_inputs)
#define NN   100000
#define NE   1600000
#define DIN  128
#define DH   128
#define DOUTD 64
#define NLAY 10

typedef float v2f __attribute__((ext_vector_type(2)));
typedef float v8f __attribute__((ext_vector_type(8)));

// CDNA5 fp32 WMMA: D(16x16,f32) = A(16x4,f32) * B(4x16,f32) + C
__device__ __forceinline__ v8f wmma4(v2f a, v2f b, v8f c) {
  return __builtin_amdgcn_wmma_f32_16x16x4_f32(false, a, false, b, (short)0, c,
                                               false, false);
}

// ---------------- degree / symmetric norm ----------------
__global__ void k_deg_init(float* __restrict__ deg) {
  int i = blockIdx.x * blockDim.x + threadIdx.x;
  if (i < NN) deg[i] = 1.0f;  // self-loop contributes 1
}
__global__ void k_deg_accum(const int* __restrict__ dst, float* __restrict__ deg) {
  int e = blockIdx.x * blockDim.x + threadIdx.x;
  if (e < NE) unsafeAtomicAdd(&deg[dst[e]], 1.0f);   // global_atomic_add_f32
}
__global__ void k_dinv(float* __restrict__ deg) {
  int i = blockIdx.x * blockDim.x + threadIdx.x;
  if (i < NN) deg[i] = rsqrtf(deg[i]);               // deg >= 1 always
}

// ---------------- WMMA GEMM: C[NN,NCOL] = A[NN,K] @ W[K,NCOL] ----------------
// One wave computes one 16x16 output tile, K-loop in steps of 4.
template<int K, int NCOL>
__global__ __launch_bounds__(128) void k_gemm(const float* __restrict__ A,
                                              const float* __restrict__ W,
                                              float* __restrict__ C) {
  const int wave = blockIdx.x * (blockDim.x >> 5) + (threadIdx.x >> 5);
  const int lane = threadIdx.x & 31;
  const int mTiles = NN / 16, nTiles = NCOL / 16;
  if (wave >= mTiles * nTiles) return;  // wave-uniform: EXEC stays all-ones
  const int mT = wave % mTiles, nT = wave / mTiles;
  const int hi  = lane >> 4;            // 0 -> K{0,1}, 1 -> K{2,3}
  const int l15 = lane & 15;
  const float* arow = A + (size_t)(mT * 16 + l15) * K;  // A row per lane (M)
  const int n = nT * 16 + l15;                          // B/C column per lane
  v8f acc = {};
#pragma unroll 4
  for (int k = 0; k < K; k += 4) {
    v2f a = *(const v2f*)(arow + k + 2 * hi);           // 8B-aligned b64 load
    v2f b;
    b.x = W[(size_t)(k + 2 * hi)     * NCOL + n];
    b.y = W[(size_t)(k + 2 * hi + 1) * NCOL + n];
    acc = wmma4(a, b, acc);
  }
  const int row0 = mT * 16 + hi * 8;    // C/D layout: lanes 16-31 hold M+8
#pragma unroll
  for (int v = 0; v < 8; ++v)
    C[(size_t)(row0 + v) * NCOL + n] = acc[v];
}

// ------------- self-loop term (also zero-initializes accumulator) -------------
template<int D>
__global__ void k_self_init(const float* __restrict__ in,
                            const float* __restrict__ dinv,
                            float* __restrict__ out) {
  size_t idx = (size_t)blockIdx.x * blockDim.x + threadIdx.x;
  const size_t total = (size_t)NN * (D / 4);
  if (idx >= total) return;
  size_t node = idx / (D / 4);
  float w = dinv[node]; w *= w;                       // norm of self-loop
  float4 v = ((const float4*)in)[idx];
  float4 r; r.x = v.x * w; r.y = v.y * w; r.z = v.z * w; r.w = v.w * w;
  ((float4*)out)[idx] = r;
}

// ------------- edge scatter-add: out[dst] += dinv[s]*dinv[d] * in[src] -------------
template<int D>
__global__ void k_scatter(const int* __restrict__ src, const int* __restrict__ dst,
                          const float* __restrict__ dinv,
                          const float* __restrict__ in, float* __restrict__ out) {
  const int TPE = D / 4;                              // threads per edge
  long long idx = (long long)blockIdx.x * blockDim.x + threadIdx.x;
  const long long total = (long long)NE * TPE;
  if (idx >= total) return;
  int e = (int)(idx / TPE);
  int c = (int)(idx % TPE) * 4;
  int s = src[e], d = dst[e];
  float w = dinv[s] * dinv[d];
  float4 v = *(const float4*)(in + (size_t)s * D + c);
  float* op = out + (size_t)d * D + c;
  unsafeAtomicAdd(op + 0, w * v.x);
  unsafeAtomicAdd(op + 1, w * v.y);
  unsafeAtomicAdd(op + 2, w * v.z);
  unsafeAtomicAdd(op + 3, w * v.w);
}

// ------------- bias + optional ReLU -------------
__global__ void k_bias_act(float* __restrict__ x, const float* __restrict__ b,
                           int D, int relu) {
  size_t idx = (size_t)blockIdx.x * blockDim.x + threadIdx.x;
  if (idx >= (size_t)NN * D) return;
  int c = (int)(idx % D);
  float v = x[idx] + b[c];
  if (relu) v = fmaxf(v, 0.0f);
  x[idx] = v;
}

// ------------- fused output: sigmoid(concat(h,xl) @ Wf + bf) (K=192) -------------
__global__ __launch_bounds__(128) void k_fused(const float* __restrict__ H,
                                               const float* __restrict__ XL,
                                               const float* __restrict__ Wf,
                                               const float* __restrict__ bf,
                                               float* __restrict__ out) {
  const int wave = blockIdx.x * (blockDim.x >> 5) + (threadIdx.x >> 5);
  const int lane = threadIdx.x & 31;
  const int mTiles = NN / 16, nTiles = DOUTD / 16;
  if (wave >= mTiles * nTiles) return;
  const int mT = wave % mTiles, nT = wave / mTiles;
  const int hi  = lane >> 4;
  const int l15 = lane & 15;
  const int n = nT * 16 + l15;
  v8f acc = {};
  const float* hrow = H + (size_t)(mT * 16 + l15) * DH;
#pragma unroll 4
  for (int k = 0; k < DH; k += 4) {                   // first 128 of K
    v2f a = *(const v2f*)(hrow + k + 2 * hi);
    v2f b;
    b.x = Wf[(size_t)(k + 2 * hi)     * DOUTD + n];
    b.y = Wf[(size_t)(k + 2 * hi + 1) * DOUTD + n];
    acc = wmma4(a, b, acc);
  }
  const float* xrow = XL + (size_t)(mT * 16 + l15) * DOUTD;
#pragma unroll 4
  for (int k = 0; k < DOUTD; k += 4) {                // last 64 of K
    v2f a = *(const v2f*)(xrow + k + 2 * hi);
    v2f b;
    b.x = Wf[(size_t)(DH + k + 2 * hi)     * DOUTD + n];
    b.y = Wf[(size_t)(DH + k + 2 * hi + 1) * DOUTD + n];
    acc = wmma4(a, b, acc);
  }
  const int row0 = mT * 16 + hi * 8;
  const float bias = bf[n];
#pragma unroll
  for (int v = 0; v < 8; ++v) {
    float z = acc[v] + bias;
    out[(size_t)(row0 + v) * DOUTD + n] = 1.0f / (1.0f + expf(-z));
  }
}

extern "C" void kernel_launch(void* const* d_in, const int* in_sizes, int n_in,
                              void* d_out, int out_size, void* d_ws, size_t ws_size,
                              hipStream_t stream) {
  (void)in_sizes; (void)n_in; (void)out_size; (void)ws_size;
  const float* x   = (const float*)d_in[0];
  const float* y   = (const float*)d_in[1];
  const int*   ei  = (const int*)  d_in[2];
  const float* Wg1 = (const float*)d_in[3];
  const float* bg1 = (const float*)d_in[4];
  const float* Wg2 = (const float*)d_in[5];
  const float* bg2 = (const float*)d_in[6];
  const float* Wl  = (const float*)d_in[7];
  const float* bl  = (const float*)d_in[8];
  const float* Wf  = (const float*)d_in[9];
  const float* bf  = (const float*)d_in[10];
  const int* src = ei;            // edge_index[0]
  const int* dst = ei + NE;       // edge_index[1]

  // Workspace layout (~154 MB): dinv | h (N*128) | g (N*128) | s (N*128)
  float* dinv = (float*)d_ws;
  float* h = dinv + NN;
  float* g = h + (size_t)NN * DH;
  float* s = g + (size_t)NN * DH;

  auto nb = [](long long t, int per) { return (unsigned)((t + per - 1) / per); };

  // Degrees -> dinv
  k_deg_init <<<nb(NN, 256), 256, 0, stream>>>(dinv);
  k_deg_accum<<<nb(NE, 256), 256, 0, stream>>>(dst, dinv);
  k_dinv     <<<nb(NN, 256), 256, 0, stream>>>(dinv);

  const long long waves128 = (long long)(NN / 16) * (DH / 16);
  const long long waves64  = (long long)(NN / 16) * (DOUTD / 16);

  // conv1: g = x@Wg1 ; s = aggregate(g) ; s = relu(s + bg1)
  k_gemm<DIN, DH><<<nb(waves128, 4), 128, 0, stream>>>(x, Wg1, g);
  k_self_init<DH><<<nb((long long)NN * (DH / 4), 256), 256, 0, stream>>>(g, dinv, s);
  k_scatter<DH>  <<<nb((long long)NE * (DH / 4), 256), 256, 0, stream>>>(src, dst, dinv, g, s);
  k_bias_act     <<<nb((long long)NN * DH, 256), 256, 0, stream>>>(s, bg1, DH, 1);

  // conv2: g = s@Wg2 ; h = aggregate(g) + bg2
  k_gemm<DH, DH><<<nb(waves128, 4), 128, 0, stream>>>(s, Wg2, g);
  k_self_init<DH><<<nb((long long)NN * (DH / 4), 256), 256, 0, stream>>>(g, dinv, h);
  k_scatter<DH>  <<<nb((long long)NE * (DH / 4), 256), 256, 0, stream>>>(src, dst, dinv, g, h);
  k_bias_act     <<<nb((long long)NN * DH, 256), 256, 0, stream>>>(h, bg2, DH, 0);

  // Label-propagation branch: reuse g (two N*64 halves) and s (first N*64)
  float* lg = g;                        // gemm output
  float* la = g + (size_t)NN * DOUTD;   // ping
  float* lb = s;                        // pong
  const float* cur = y;
  float* nxt = la;
  for (int j = 0; j < NLAY; ++j) {
    k_gemm<DOUTD, DOUTD><<<nb(waves64, 4), 128, 0, stream>>>(
        cur, Wl + (size_t)j * DOUTD * DOUTD, lg);
    k_self_init<DOUTD><<<nb((long long)NN * (DOUTD / 4), 256), 256, 0, stream>>>(lg, dinv, nxt);
    k_scatter<DOUTD>  <<<nb((long long)NE * (DOUTD / 4), 256), 256, 0, stream>>>(src, dst, dinv, lg, nxt);
    k_bias_act        <<<nb((long long)NN * DOUTD, 256), 256, 0, stream>>>(
        nxt, bl + (size_t)j * DOUTD, DOUTD, (j != NLAY - 1) ? 1 : 0);
    cur = nxt;
    nxt = (nxt == la) ? lb : la;
  }

  // Fused head: sigmoid([h | xl] @ Wf + bf) -> d_out
  k_fused<<<nb(waves64, 4), 128, 0, stream>>>(h, cur, Wf, bf, (float*)d_out);
}